// PathDecoder_12120397710138
// MI455X (gfx1250) — compile-verified
//
#include <hip/hip_runtime.h>
#include <hip/hip_bf16.h>

// Problem constants (match reference)
#define B_   32
#define N_   10000
#define G_   32
#define H_   128
#define NH_  8
#define HD_  16
#define KNN_ 64
#define HP_  132          // LDS row pitch (floats): 528B = 16B aligned, bank-safe
#define NCHUNK_ 25        // mean-reduction chunks (25 * 400 = 10000)

typedef float v2f __attribute__((ext_vector_type(2)));
typedef float v4f __attribute__((ext_vector_type(4)));
typedef float v8f __attribute__((ext_vector_type(8)));
typedef unsigned int u32x4 __attribute__((ext_vector_type(4)));
typedef int i32x4 __attribute__((ext_vector_type(4)));
typedef int i32x8 __attribute__((ext_vector_type(8)));

static __device__ __forceinline__ v8f wmma_f32x4(v2f a, v2f b, v8f c) {
  // D = A(16x4 f32) * B(4x16 f32) + C(16x16 f32)   -> v_wmma_f32_16x16x4_f32
  return __builtin_amdgcn_wmma_f32_16x16x4_f32(false, a, false, b, (short)0, c,
                                               false, false);
}

// ---------------------------------------------------------------------------
// K1a: partial sums of embeddings over N (deterministic two-stage mean).
// Each block produces 4 partial rows (one per 32-thread row group), float4 IO.
// grid (B_, NCHUNK_), block 128
__global__ __launch_bounds__(128) void k_mean_part(const float* __restrict__ emb,
                                                   float* __restrict__ part) {
  int b = blockIdx.x, chunk = blockIdx.y, t = threadIdx.x;
  int rr = t >> 5, q = t & 31;
  const int span = N_ / NCHUNK_; // 400
  const float* base = emb + ((size_t)b * N_ + (size_t)chunk * span + rr) * H_ + q * 4;
  v4f s = {0.f, 0.f, 0.f, 0.f};
  for (int n = 0; n < span; n += 4) s += *(const v4f*)(base + (size_t)n * H_);
  *(v4f*)&part[(((size_t)b * NCHUNK_ + chunk) * 4 + rr) * H_ + q * 4] = s;
}

// K1b: finalize mean over 100 partial rows. grid B_, block 128
__global__ __launch_bounds__(128) void k_mean_fin(const float* __restrict__ part,
                                                  float* __restrict__ meanv) {
  int b = blockIdx.x, h = threadIdx.x;
  float s = 0.f;
  for (int c = 0; c < NCHUNK_ * 4; ++c)
    s += part[((size_t)b * NCHUNK_ * 4 + c) * H_ + h];
  meanv[b * H_ + h] = s * (1.0f / N_);
}

// ---------------------------------------------------------------------------
// K2: glimpse_q[m, :] = mean_b@Wq_graph + e[src]@Wq_source + e[tgt]@Wq_target
//                      + e[first]@Wq_first + e[last]@Wq_last     (m = b*G+g)
// 1024x128 output, five accumulated GEMMs via WMMA f32 16x16x4.
// grid 64 (m-tiles), block 256 (8 waves = 8 n-tiles)
__global__ __launch_bounds__(256) void k_gq(
    const float* __restrict__ meanv, const float* __restrict__ emb,
    const int* __restrict__ srcn, const int* __restrict__ tgtn,
    const int* __restrict__ fstn, const int* __restrict__ lstn,
    const float* __restrict__ Wg, const float* __restrict__ Ws,
    const float* __restrict__ Wt, const float* __restrict__ Wf,
    const float* __restrict__ Wl, float* __restrict__ gq) {
  int t = threadIdx.x, lane = t & 31, wid = t >> 5;
  int mt = blockIdx.x;
  int n0 = wid * 16;
  int r = lane & 15;               // A row within tile
  int kh = (lane >> 4) << 1;       // K sub-offset: lanes 16-31 hold K=2,3
  int nn = lane & 15;              // B col
  int m = mt * 16 + r;
  int b = m / G_;

  const float* rows[5];
  rows[0] = meanv + (size_t)b * H_;
  rows[1] = emb + ((size_t)b * N_ + srcn[m]) * H_;
  rows[2] = emb + ((size_t)b * N_ + tgtn[m]) * H_;
  rows[3] = emb + ((size_t)b * N_ + fstn[m]) * H_;
  rows[4] = emb + ((size_t)b * N_ + lstn[m]) * H_;
  const float* Wmat[5] = {Wg, Ws, Wt, Wf, Wl};

  v8f acc = {};
  for (int q = 0; q < 5; ++q) {
    const float* rp = rows[q];
    const float* W = Wmat[q];
    for (int k = 0; k < H_; k += 4) {
      v2f a = *(const v2f*)(rp + k + kh);
      v2f bb; bb.x = W[(size_t)(k + kh) * H_ + n0 + nn];
              bb.y = W[(size_t)(k + kh + 1) * H_ + n0 + nn];
      acc = wmma_f32x4(a, bb, acc);
    }
  }
  int ro = (lane >> 4) * 8;
  for (int v = 0; v < 8; ++v)
    gq[(size_t)(mt * 16 + ro + v) * H_ + n0 + nn] = acc[v];
}

// ---------------------------------------------------------------------------
// K3: KNN top-64 smallest squared distances per (b,g).
// Binary-search threshold over LDS-resident distances, deterministic
// prefix-sum compaction (no float atomics). grid B_*G_, block 256.
__global__ __launch_bounds__(256) void k_knn(const float* __restrict__ coords,
                                             const float* __restrict__ mask,
                                             const int* __restrict__ lastn,
                                             int* __restrict__ knn) {
  __shared__ float sd[N_];       // 40 KB
  __shared__ int   ibuf[257];
  int bg = blockIdx.x, b = bg >> 5, t = threadIdx.x;
  int last = lastn[bg];
  const float* cb = coords + (size_t)b * N_ * 2;
  const float* mb = mask + (size_t)bg * N_;
  v2f lc = *(const v2f*)(cb + (size_t)last * 2);

  for (int n = t; n < N_; n += 256) {
    v2f c = *(const v2f*)(cb + (size_t)n * 2);
    float dx = lc.x - c.x, dy = lc.y - c.y;
    float d = dx * dx + dy * dy;
    if (mb[n] < -1e37f) d = __builtin_inff();  // masked-out -> +inf
    sd[n] = d;
  }
  __syncthreads();

  // binary search for threshold: cnt(lo) < KNN <= cnt(hi), cnt(x)=#{d<x}
  float lo = 0.0f, hi = 4.0f;
  for (int it = 0; it < 32; ++it) {
    float mid = 0.5f * (lo + hi);
    int c = 0;
    for (int n = t; n < N_; n += 256) c += (sd[n] < mid);
    ibuf[t] = c; __syncthreads();
    for (int s = 128; s > 0; s >>= 1) { if (t < s) ibuf[t] += ibuf[t + s]; __syncthreads(); }
    int total = ibuf[0]; __syncthreads();
    if (total >= KNN_) hi = mid; else lo = mid;
  }

  // pass A: d < lo  (count < KNN)
  int my = 0;
  for (int n = t; n < N_; n += 256) my += (sd[n] < lo);
  ibuf[t] = my; __syncthreads();
  if (t == 0) { int run = 0; for (int i = 0; i < 256; ++i) { int c = ibuf[i]; ibuf[i] = run; run += c; } ibuf[256] = run; }
  __syncthreads();
  int base = ibuf[t], totalA = ibuf[256];
  for (int n = t; n < N_; n += 256)
    if (sd[n] < lo) { if (base < KNN_) knn[(size_t)bg * KNN_ + base] = n; ++base; }
  __syncthreads();

  // pass B: ties at the threshold, fill up to KNN
  my = 0;
  for (int n = t; n < N_; n += 256) my += (sd[n] >= lo && sd[n] < hi);
  ibuf[t] = my; __syncthreads();
  if (t == 0) { int run = 0; for (int i = 0; i < 256; ++i) { int c = ibuf[i]; ibuf[i] = run; run += c; } ibuf[256] = run; }
  __syncthreads();
  int baseB = totalA + ibuf[t], total = totalA + ibuf[256];
  for (int n = t; n < N_; n += 256)
    if (sd[n] >= lo && sd[n] < hi) { if (baseB < KNN_) knn[(size_t)bg * KNN_ + baseB] = n; ++baseB; }
  __syncthreads();
  if (t == 0) for (int p = total; p < KNN_; ++p) knn[(size_t)bg * KNN_ + p] = last;
}

// ---------------------------------------------------------------------------
// K4: sparse glimpse attention + combine, per (b,g).
// Gather 64 neighbor embeddings -> LDS; K=E@Wk, V=E@Wv via WMMA into LDS;
// 8-head softmax over 64; heads_out@W_comb + b_comb -> final_q.
// grid B_*G_, block 256, dynamic LDS.
__global__ __launch_bounds__(256) void k_attn(
    const float* __restrict__ emb, const float* __restrict__ Wk,
    const float* __restrict__ Wv, const float* __restrict__ Wcomb,
    const float* __restrict__ bcomb, const float* __restrict__ gq,
    const int* __restrict__ knn, float* __restrict__ fq) {
  extern __shared__ float smem[];
  float* sE = smem;                      // KNN_*HP_
  float* sK = sE + KNN_ * HP_;           // KNN_*HP_
  float* sV = sK + KNN_ * HP_;           // KNN_*HP_
  float* sS = sV + KNN_ * HP_;           // NH_*KNN_
  float* sHO = sS + NH_ * KNN_;          // H_

  int bg = blockIdx.x, b = bg >> 5;
  int t = threadIdx.x, lane = t & 31, wid = t >> 5;
  const int* nb = knn + (size_t)bg * KNN_;

  // gather neighbor embeddings, 16B granules (coalesced per row)
  for (int idx = t; idx < KNN_ * (H_ / 4); idx += 256) {
    int r = idx >> 5, c4 = idx & 31;
    *(v4f*)&sE[r * HP_ + c4 * 4] =
        *(const v4f*)&emb[((size_t)b * N_ + nb[r]) * H_ + c4 * 4];
  }
  __syncthreads();

  // K (waves 0-3) / V (waves 4-7) transform via WMMA
  {
    const float* W = (wid < 4) ? Wk : Wv;
    float* dst = (wid < 4) ? sK : sV;
    int mt = wid & 3;
    int r = lane & 15, kh = (lane >> 4) << 1, nn = lane & 15;
    int ro = (lane >> 4) * 8;
    for (int nt = 0; nt < 8; ++nt) {
      v8f acc = {};
      for (int k = 0; k < H_; k += 4) {
        v2f a = *(const v2f*)&sE[(mt * 16 + r) * HP_ + k + kh];
        v2f bb; bb.x = W[(size_t)(k + kh) * H_ + nt * 16 + nn];
                bb.y = W[(size_t)(k + kh + 1) * H_ + nt * 16 + nn];
        acc = wmma_f32x4(a, bb, acc);
      }
      for (int v = 0; v < 8; ++v)
        dst[(mt * 16 + ro + v) * HP_ + nt * 16 + nn] = acc[v];
    }
  }
  __syncthreads();

  // scores s[nh][j] = (q_h . k_j) / sqrt(HD)
  const float* q = gq + (size_t)bg * H_;
  for (int p = t; p < NH_ * KNN_; p += 256) {
    int nh = p >> 6, j = p & 63;
    float s = 0.f;
    for (int d = 0; d < HD_; ++d) s += q[nh * HD_ + d] * sK[j * HP_ + nh * HD_ + d];
    sS[nh * KNN_ + j] = s * 0.25f;  // 1/sqrt(16)
  }
  __syncthreads();

  // per-head softmax over the 64 neighbors
  if (t < NH_) {
    float m = -__builtin_inff();
    for (int j = 0; j < KNN_; ++j) m = fmaxf(m, sS[t * KNN_ + j]);
    float sum = 0.f;
    for (int j = 0; j < KNN_; ++j) { float e = __expf(sS[t * KNN_ + j] - m); sS[t * KNN_ + j] = e; sum += e; }
    float inv = 1.0f / sum;
    for (int j = 0; j < KNN_; ++j) sS[t * KNN_ + j] *= inv;
  }
  __syncthreads();

  // heads_out
  if (t < H_) {
    int nh = t >> 4;
    float a = 0.f;
    for (int j = 0; j < KNN_; ++j) a += sS[nh * KNN_ + j] * sV[j * HP_ + t];
    sHO[t] = a;
  }
  __syncthreads();

  // final_q = heads_out @ W_comb + b_comb
  if (t < H_) {
    float a = bcomb[t];
    for (int hi = 0; hi < H_; ++hi) a += sHO[hi] * Wcomb[(size_t)hi * H_ + t];
    fq[(size_t)bg * H_ + t] = a;
  }
}

// ---------------------------------------------------------------------------
// K5a: pointer logits  out[b,g,n] = 10*tanh((fq . emb_n)/sqrt(H)) + mask
// WMMA GEMM per batch: [32 x 128] @ [128 x 128-chunk]. The 128x128 f32
// embedding tile is DMA'd to LDS by the Tensor Data Mover with hardware row
// padding (4 DWORDs per 128 -> 132-float pitch). grid (B_, 79), block 256.
__global__ __launch_bounds__(256) void k_logits(const float* __restrict__ emb,
                                                const float* __restrict__ fq,
                                                const float* __restrict__ mask,
                                                float* __restrict__ out) {
  extern __shared__ float smem[];
  float* sEmb = smem;               // 128 * HP_
  float* sFQ = sEmb + 128 * HP_;    // G_ * HP_

  int b = blockIdx.x;
  int n0 = blockIdx.y * 128;
  int t = threadIdx.x, lane = t & 31, wid = t >> 5;

#if __has_builtin(__builtin_amdgcn_tensor_load_to_lds)
  // --- TDM path: one DMA per block, issued by wave 0 (EXEC ignored by TDM) ---
  if (wid == 0) {
    int rows = N_ - n0; if (rows > 128) rows = 128;   // OOB rows read as zero
    unsigned long long ga =
        (unsigned long long)(uintptr_t)(emb + ((size_t)b * N_ + n0) * H_);
    unsigned lds_base = (unsigned)(uintptr_t)sEmb;    // low 32 bits = LDS offset
    u32x4 g0;
    g0.x = 1u;                                        // count=1, no gather
    g0.y = lds_base;                                  // lds_addr
    g0.z = (unsigned)ga;                              // global_addr[31:0]
    g0.w = (unsigned)((ga >> 32) & 0x1FFFFFFu) | (2u << 30);  // addr msbs, type=2
    i32x8 g1;
    g1.s0 = (2 << 16)      // data_size = 4B
          | (1 << 20)      // pad_enable
          | (6 << 22)      // pad_interval: every 128 DWORDs
          | (3 << 25);     // pad_amount: 4 DWORDs  -> pitch 132 floats
    g1.s1 = (H_ & 0xFFFF) << 16;          // tensor_dim0 = 128 (bits 63:48)
    g1.s2 = (rows & 0xFFFF) << 16;        // tensor_dim1 (bits 95:80)
    g1.s3 = (128 << 16);                  // tile_dim0 = 128 (bits 127:112)
    g1.s4 = 128;                          // tile_dim1 = 128, tile_dim2 = 0
    g1.s5 = H_;                           // tensor_dim0_stride = 128 elements
    g1.s6 = 0; g1.s7 = 0;                 // dim1 stride unused (2D tile)
    i32x4 zz = {0, 0, 0, 0};
#if __clang_major__ >= 23
    i32x8 z8 = {0, 0, 0, 0, 0, 0, 0, 0};
    __builtin_amdgcn_tensor_load_to_lds(g0, g1, zz, zz, z8, 0);
#else
    __builtin_amdgcn_tensor_load_to_lds(g0, g1, zz, zz, 0);
#endif
    __builtin_amdgcn_s_wait_tensorcnt((short)0);
  }
#else
  // --- fallback: manual staged copy ---
  for (int idx = t; idx < 128 * (H_ / 4); idx += 256) {
    int rr = idx >> 5, c4 = idx & 31;
    int n = n0 + rr;
    v4f val = {0.f, 0.f, 0.f, 0.f};
    if (n < N_) val = *(const v4f*)&emb[((size_t)b * N_ + n) * H_ + c4 * 4];
    *(v4f*)&sEmb[rr * HP_ + c4 * 4] = val;
  }
#endif
  for (int idx = t; idx < G_ * (H_ / 4); idx += 256) {
    int rr = idx >> 5, c4 = idx & 31;
    *(v4f*)&sFQ[rr * HP_ + c4 * 4] =
        *(const v4f*)&fq[((size_t)b * G_ + rr) * H_ + c4 * 4];
  }
  __syncthreads();

  int nt = wid;
  int r = lane & 15, kh = (lane >> 4) << 1, nn = lane & 15;
  int n = n0 + nt * 16 + nn;
  int ro = (lane >> 4) * 8;
  for (int mt = 0; mt < 2; ++mt) {
    v8f acc = {};
    for (int k = 0; k < H_; k += 4) {
      v2f a = *(const v2f*)&sFQ[(mt * 16 + r) * HP_ + k + kh];
      v2f bb = *(const v2f*)&sEmb[(nt * 16 + nn) * HP_ + k + kh];
      acc = wmma_f32x4(a, bb, acc);
    }
    if (n < N_) {
      for (int v = 0; v < 8; ++v) {
        int g = mt * 16 + ro + v;
        float s = acc[v] * 0.08838834764831845f;  // 1/sqrt(128)
        size_t o = ((size_t)(b * G_ + g)) * N_ + n;
        out[o] = 10.0f * tanhf(s) + mask[o];
      }
    }
  }
}

// ---------------------------------------------------------------------------
// K5b: softmax over N per (b,g) row, in place in d_out. grid B_*G_, block 256.
__global__ __launch_bounds__(256) void k_softmax(float* __restrict__ out) {
  __shared__ float sbuf[256];
  int bg = blockIdx.x, t = threadIdx.x;
  float* row = out + (size_t)bg * N_;

  float m = -__builtin_inff();
  for (int n = t; n < N_; n += 256) m = fmaxf(m, row[n]);
  sbuf[t] = m; __syncthreads();
  for (int s = 128; s > 0; s >>= 1) { if (t < s) sbuf[t] = fmaxf(sbuf[t], sbuf[t + s]); __syncthreads(); }
  m = sbuf[0]; __syncthreads();

  float sum = 0.f;
  for (int n = t; n < N_; n += 256) sum += __expf(row[n] - m);
  sbuf[t] = sum; __syncthreads();
  for (int s = 128; s > 0; s >>= 1) { if (t < s) sbuf[t] += sbuf[t + s]; __syncthreads(); }
  float inv = 1.0f / sbuf[0]; __syncthreads();

  for (int n = t; n < N_; n += 256) row[n] = __expf(row[n] - m) * inv;
}

// ---------------------------------------------------------------------------
extern "C" void kernel_launch(void* const* d_in, const int* in_sizes, int n_in,
                              void* d_out, int out_size, void* d_ws, size_t ws_size,
                              hipStream_t stream) {
  const float* coords = (const float*)d_in[0];
  const float* emb    = (const float*)d_in[1];
  const float* mask   = (const float*)d_in[2];
  const int*   srcn   = (const int*)d_in[3];
  const int*   tgtn   = (const int*)d_in[4];
  const int*   fstn   = (const int*)d_in[5];
  const int*   lstn   = (const int*)d_in[6];
  const float* Wg     = (const float*)d_in[7];
  const float* Ws     = (const float*)d_in[8];
  const float* Wt     = (const float*)d_in[9];
  const float* Wf     = (const float*)d_in[10];
  const float* Wl     = (const float*)d_in[11];
  const float* Wk     = (const float*)d_in[12];
  const float* Wv     = (const float*)d_in[13];
  const float* Wcomb  = (const float*)d_in[14];
  const float* bcomb  = (const float*)d_in[15];
  float* out = (float*)d_out;

  // workspace partition (floats): part | mean | gq | fq | knn(int)
  float* ws_part = (float*)d_ws;                       // B*NCHUNK*4*H = 409600
  float* ws_mean = ws_part + B_ * NCHUNK_ * 4 * H_;    // B*H          = 4096
  float* ws_gq   = ws_mean + B_ * H_;                  // B*G*H        = 131072
  float* ws_fq   = ws_gq + B_ * G_ * H_;               // B*G*H        = 131072
  int*   ws_knn  = (int*)(ws_fq + B_ * G_ * H_);       // B*G*KNN      = 65536 ints

  k_mean_part<<<dim3(B_, NCHUNK_), 128, 0, stream>>>(emb, ws_part);
  k_mean_fin<<<B_, 128, 0, stream>>>(ws_part, ws_mean);

  k_gq<<<(B_ * G_) / 16, 256, 0, stream>>>(ws_mean, emb, srcn, tgtn, fstn, lstn,
                                           Wg, Ws, Wt, Wf, Wl, ws_gq);

  k_knn<<<B_ * G_, 256, 0, stream>>>(coords, mask, lstn, ws_knn);

  size_t attn_lds = (size_t)(3 * KNN_ * HP_ + NH_ * KNN_ + H_) * sizeof(float);
  k_attn<<<B_ * G_, 256, attn_lds, stream>>>(emb, Wk, Wv, Wcomb, bcomb, ws_gq,
                                             ws_knn, ws_fq);

  size_t log_lds = (size_t)(128 * HP_ + G_ * HP_) * sizeof(float);
  k_logits<<<dim3(B_, (N_ + 127) / 128), 256, log_lds, stream>>>(emb, ws_fq, mask, out);

  k_softmax<<<B_ * G_, 256, 0, stream>>>(out);
}